// ReadHead_53137335386348
// MI455X (gfx1250) — compile-verified
//
#include <hip/hip_runtime.h>
#include <math.h>

typedef float v2f __attribute__((ext_vector_type(2)));
typedef float v8f __attribute__((ext_vector_type(8)));

#define B_    8192
#define C_    256
#define N_    128
#define M_    20
#define S_    3
#define EPSF  1e-8f
#define PCOLS 32   // 26 head params padded to 32

// ---------------------------------------------------------------------------
// Kernel 1: pack [Wk | Wb | Wg | Ws | Wgam] into a [256 x 32] panel (pad -> 0)
// ---------------------------------------------------------------------------
__global__ void pack_weights(const float* __restrict__ Wk, const float* __restrict__ Wb,
                             const float* __restrict__ Wg, const float* __restrict__ Ws,
                             const float* __restrict__ Wgam, float* __restrict__ Wp) {
    int t = threadIdx.x;              // row of C (0..255)
    float* row = Wp + t * PCOLS;
    for (int j = 0; j < M_; ++j) row[j] = Wk[t * M_ + j];
    row[20] = Wb[t];
    row[21] = Wg[t];
    row[22] = Ws[t * S_ + 0];
    row[23] = Ws[t * S_ + 1];
    row[24] = Ws[t * S_ + 2];
    row[25] = Wgam[t];
    for (int j = 26; j < PCOLS; ++j) row[j] = 0.0f;
}

// ---------------------------------------------------------------------------
// Kernel 2: P[B x 32] = co[B x 256] @ Wp[256 x 32] using V_WMMA_F32_16X16X4_F32
// block = 256 threads = 8 waves, each wave owns a 16-row (batch) x 32-col tile
// ---------------------------------------------------------------------------
__global__ void head_gemm(const float* __restrict__ co, const float* __restrict__ Wp,
                          float* __restrict__ P) {
    __shared__ float sW[C_ * PCOLS];  // 32 KB weight panel shared by 8 waves

    int tid = threadIdx.x;
    for (int i = tid; i < C_ * PCOLS; i += 256) sW[i] = Wp[i];
    __syncthreads();

    const int wave = tid >> 5;
    const int lane = tid & 31;
    const int hi   = lane >> 4;       // 0: K pair {0,1}, 1: K pair {2,3}
    const int m    = lane & 15;       // row (A) / col (B,C)
    const int b0   = (blockIdx.x * 8 + wave) * 16;

    v8f acc0 = {};                    // cols 0..15
    v8f acc1 = {};                    // cols 16..31

    for (int k0 = 0; k0 < C_; k0 += 4) {
        const int ka = k0 + 2 * hi;
        // A fragment: 16x4 f32
        v2f a;
        a.x = co[(size_t)(b0 + m) * C_ + ka];
        a.y = co[(size_t)(b0 + m) * C_ + ka + 1];
        // B fragments: 4x16 f32 (two col tiles)
        v2f bA, bB;
        bA.x = sW[ka * PCOLS + m];
        bA.y = sW[(ka + 1) * PCOLS + m];
        bB.x = sW[ka * PCOLS + 16 + m];
        bB.y = sW[(ka + 1) * PCOLS + 16 + m];
        acc0 = __builtin_amdgcn_wmma_f32_16x16x4_f32(false, a, false, bA, (short)0, acc0, false, false);
        acc1 = __builtin_amdgcn_wmma_f32_16x16x4_f32(false, a, false, bB, (short)0, acc1, false, false);
    }

    // C/D layout: VGPR v -> row v (lanes 0-15) / v+8 (lanes 16-31), col = lane%16
    for (int v = 0; v < 8; ++v) {
        const int row = v + 8 * hi;
        P[(size_t)(b0 + row) * PCOLS + m]      = acc0[v];
        P[(size_t)(b0 + row) * PCOLS + 16 + m] = acc1[v];
    }
}

// ---------------------------------------------------------------------------
// Kernel 3: fused NTM addressing, one block (128 threads) per batch element
// ---------------------------------------------------------------------------
__device__ __forceinline__ float softplusf(float x) {
    return fmaxf(x, 0.0f) + log1pf(expf(-fabsf(x)));
}

__global__ void ntm_head(const float* __restrict__ P,
                         const float* __restrict__ bk, const float* __restrict__ bb,
                         const float* __restrict__ bg, const float* __restrict__ bs,
                         const float* __restrict__ bgam,
                         const float* __restrict__ w_prev,
                         const float* __restrict__ Mt,
                         float* __restrict__ r_out, float* __restrict__ w_out) {
    __shared__ float sP[PCOLS];
    __shared__ float sK[M_];
    __shared__ float sRed[N_];
    __shared__ float sWg[N_];
    __shared__ float sWt[N_];
    __shared__ float sBeta, sG, sGamma, sS0, sS1, sS2, sKn;

    const int tid = threadIdx.x;      // 0..127 == memory location n
    const int b   = blockIdx.x;

    // ---- head params: add bias, stage raw P row ----
    if (tid < 26) {
        float bias;
        if      (tid < 20)  bias = bk[tid];
        else if (tid == 20) bias = bb[0];
        else if (tid == 21) bias = bg[0];
        else if (tid < 25)  bias = bs[tid - 22];
        else                bias = bgam[0];
        sP[tid] = P[(size_t)b * PCOLS + tid] + bias;
    }
    __syncthreads();

    if (tid < 20) {
        sK[tid] = tanhf(sP[tid]);
    } else if (tid == 20) {
        sBeta = softplusf(sP[20]);
    } else if (tid == 21) {
        sG = 1.0f / (1.0f + expf(-sP[21]));
    } else if (tid == 22) {
        sGamma = 1.0f + softplusf(sP[25]);
    } else if (tid == 23) {
        float a = sP[22], c = sP[23], d = sP[24];
        float mx = fmaxf(a, fmaxf(c, d));
        float e0 = expf(a - mx), e1 = expf(c - mx), e2 = expf(d - mx);
        float s = e0 + e1 + e2;
        sS0 = e0 / s; sS1 = e1 / s; sS2 = e2 / s;
    }
    __syncthreads();

    if (tid == 0) {
        float s = 0.0f;
        for (int m = 0; m < M_; ++m) s += sK[m] * sK[m];
        sKn = sqrtf(s);
    }

    // ---- stream this location's memory row (5x b128 loads, coalesced) ----
    float mr[M_];
    const float* mrow = Mt + ((size_t)b * N_ + tid) * M_;
    #pragma unroll
    for (int m = 0; m < M_; m += 4) {
        float4 v = *(const float4*)(mrow + m);
        mr[m] = v.x; mr[m + 1] = v.y; mr[m + 2] = v.z; mr[m + 3] = v.w;
    }
    float dotv = 0.0f, mn2 = 0.0f;
    #pragma unroll
    for (int m = 0; m < M_; ++m) {
        dotv += mr[m] * sK[m];
        mn2  += mr[m] * mr[m];
    }
    __syncthreads();   // sKn visible

    // ---- content addressing: softmax(beta * cosine) over N=128 ----
    const float cosv = dotv / (sKn * sqrtf(mn2) + EPSF);
    const float aval = sBeta * cosv;

    sRed[tid] = aval;
    __syncthreads();
    for (int off = N_ / 2; off > 0; off >>= 1) {
        if (tid < off) sRed[tid] = fmaxf(sRed[tid], sRed[tid + off]);
        __syncthreads();
    }
    const float mx = sRed[0];
    __syncthreads();

    const float e = expf(aval - mx);
    sRed[tid] = e;
    __syncthreads();
    for (int off = N_ / 2; off > 0; off >>= 1) {
        if (tid < off) sRed[tid] += sRed[tid + off];
        __syncthreads();
    }
    const float wc = e / sRed[0];
    __syncthreads();

    // ---- gate + circular shift ----
    const float wg = sG * wc + (1.0f - sG) * w_prev[(size_t)b * N_ + tid];
    sWg[tid] = wg;
    __syncthreads();

    // shifts (1,0,-1): w~(i) = s0*wg[i-1] + s1*wg[i] + s2*wg[i+1]  (mod N)
    const float wtild = sS0 * sWg[(tid + N_ - 1) & (N_ - 1)]
                      + sS1 * sWg[tid]
                      + sS2 * sWg[(tid + 1) & (N_ - 1)];

    // ---- sharpen + normalize ----
    const float wpow = powf(wtild, sGamma);
    sRed[tid] = wpow;
    __syncthreads();
    for (int off = N_ / 2; off > 0; off >>= 1) {
        if (tid < off) sRed[tid] += sRed[tid + off];
        __syncthreads();
    }
    const float wt = wpow / (sRed[0] + EPSF);
    sWt[tid] = wt;
    w_out[(size_t)b * N_ + tid] = wt;
    __syncthreads();

    // ---- weighted read r[m] = sum_n w[n] * M[n][m] (rows hot in L0/L2) ----
    if (tid < M_) {
        float r = 0.0f;
        const float* base = Mt + (size_t)b * N_ * M_ + tid;
        for (int n = 0; n < N_; ++n) r += sWt[n] * base[(size_t)n * M_];
        r_out[(size_t)b * M_ + tid] = r;
    }
}

// ---------------------------------------------------------------------------
extern "C" void kernel_launch(void* const* d_in, const int* in_sizes, int n_in,
                              void* d_out, int out_size, void* d_ws, size_t ws_size,
                              hipStream_t stream) {
    const float* co     = (const float*)d_in[0];
    const float* w_prev = (const float*)d_in[1];
    const float* Mt     = (const float*)d_in[2];
    const float* Wk     = (const float*)d_in[3];
    const float* bk     = (const float*)d_in[4];
    const float* Wb     = (const float*)d_in[5];
    const float* bb     = (const float*)d_in[6];
    const float* Wg     = (const float*)d_in[7];
    const float* bg     = (const float*)d_in[8];
    const float* Ws     = (const float*)d_in[9];
    const float* bs     = (const float*)d_in[10];
    const float* Wgam   = (const float*)d_in[11];
    const float* bgam   = (const float*)d_in[12];

    float* out   = (float*)d_out;
    float* r_out = out;                         // [B, M]
    float* w_out = out + (size_t)B_ * M_;       // [B, N]

    float* Wp = (float*)d_ws;                   // 256*32 floats
    float* P  = Wp + C_ * PCOLS;                // 8192*32 floats (~1 MB total)

    pack_weights<<<1, 256, 0, stream>>>(Wk, Wb, Wg, Ws, Wgam, Wp);
    head_gemm<<<B_ / (16 * 8), 256, 0, stream>>>(co, Wp, P);
    ntm_head<<<B_, 128, 0, stream>>>(P, bk, bb, bg, bs, bgam, w_prev, Mt, r_out, w_out);
}